// MMLGroundedCoreferencer_76270029242837
// MI455X (gfx1250) — compile-verified
//
#include <hip/hip_runtime.h>

typedef float v2f __attribute__((ext_vector_type(2)));
typedef float v8f __attribute__((ext_vector_type(8)));

#define NSAMP 64
#define NS 32
#define NV 36
#define D  512
#define NVP 48    // NV padded to 3 WMMA M/N tiles
#define ATT_P 40  // att row pitch = K-padding for matmul2 (multiple of 4)

// dynamic LDS layout (float offsets)
#define OFF_FIRST  0
#define OFF_SECOND (OFF_FIRST + NS * D)        // 48x512, rows 36..47 zero
#define OFF_ATT    (OFF_SECOND + NVP * D)      // 32x40 raw att
#define OFF_AWA    (OFF_ATT + NS * ATT_P)      // 32x40 row-softmax, cols 36..39 zero
#define OFF_AWB    (OFF_AWA + NS * ATT_P)      // 48x32 col-softmax, rows 36..47 zero
#define OFF_SMASK  (OFF_AWB + NVP * NS)        // 32
#define OFF_IMASK  (OFF_SMASK + NS)            // 36
#define OFF_RED    (OFF_IMASK + NV)            // 256
#define LDS_FLOATS (OFF_RED + 256)

__global__ __launch_bounds__(256, 1)
void pair_score_kernel(const float* __restrict__ span,    // (64,32,512)
                       const float* __restrict__ image,   // (64,36,512)
                       const float* __restrict__ smaskg,  // (64,32)
                       const float* __restrict__ imaskg,  // (64,36)
                       float* __restrict__ S)             // (64,64)
{
    extern __shared__ float lds[];
    const int s    = blockIdx.x;
    const int v    = blockIdx.y;
    const int tid  = threadIdx.x;
    const int lane = tid & 31;
    const int wave = tid >> 5;       // 0..7
    const int lh   = lane >> 4;      // lane half (0/1)
    const int lm   = lane & 15;

    float* firstL  = lds + OFF_FIRST;
    float* secondL = lds + OFF_SECOND;
    float* attL    = lds + OFF_ATT;
    float* awA     = lds + OFF_AWA;
    float* awB     = lds + OFF_AWB;
    float* smask   = lds + OFF_SMASK;
    float* imask   = lds + OFF_IMASK;
    float* red     = lds + OFF_RED;

    // ---- stage inputs to LDS (coalesced float4) + zero pads ----
    {
        const float4* gf = (const float4*)(span + (size_t)s * NS * D);
        float4* lf = (float4*)firstL;
        for (int i = tid; i < NS * D / 4; i += blockDim.x) lf[i] = gf[i];

        const float4* gi = (const float4*)(image + (size_t)v * NV * D);
        float4* lsv = (float4*)secondL;
        for (int i = tid; i < NV * D / 4; i += blockDim.x) lsv[i] = gi[i];
        float4 z4 = make_float4(0.f, 0.f, 0.f, 0.f);
        for (int i = tid; i < (NVP - NV) * D / 4; i += blockDim.x) lsv[NV * D / 4 + i] = z4;

        for (int i = tid; i < NS * ATT_P; i += blockDim.x) awA[i] = 0.f;
        for (int i = tid; i < NVP * NS; i += blockDim.x)  awB[i] = 0.f;
        if (tid < NS) smask[tid] = smaskg[s * NS + tid];
        if (tid < NV) imask[tid] = imaskg[v * NV + tid];
    }
    __syncthreads();

    // ---- matmul 1: att(32x48) = first(32x512) @ second^T ----
    if (wave < 6) {
        const int mt = wave / 3;          // 0..1
        const int nt = wave % 3;          // 0..2
        v8f c = {};
        const float* arow = firstL  + (mt * 16 + lm) * D;   // A: row=lm, K pair by lane half
        const float* brow = secondL + (nt * 16 + lm) * D;   // B = second^T: col n -> second row n
        for (int k = 0; k < D; k += 4) {
            const int kk = k + 2 * lh;
            v2f a = *(const v2f*)(arow + kk);
            v2f b = *(const v2f*)(brow + kk);
            c = __builtin_amdgcn_wmma_f32_16x16x4_f32(false, a, false, b, (short)0, c, false, false);
        }
        const int col = nt * 16 + lm;
        if (col < NV) {
#pragma unroll
            for (int i = 0; i < 8; ++i) {
                const int row = mt * 16 + lh * 8 + i;
                attL[row * ATT_P + col] = c[i];
            }
        }
    }
    __syncthreads();

    // ---- softmaxes with multiplicative masks (as in reference) ----
    if (tid < NS) {  // aw_first: softmax over NV per span row
        const int r = tid;
        float mx = -3.4e38f;
        for (int j = 0; j < NV; ++j) mx = fmaxf(mx, attL[r * ATT_P + j] * imask[j]);
        float sum = 0.f;
        for (int j = 0; j < NV; ++j) {
            float e = expf(attL[r * ATT_P + j] * imask[j] - mx);
            awA[r * ATT_P + j] = e;
            sum += e;
        }
        const float inv = 1.f / sum;
        for (int j = 0; j < NV; ++j) awA[r * ATT_P + j] *= inv;
    }
    if (tid >= 64 && tid < 64 + NV) {  // aw_second: softmax over NS per image row
        const int r = tid - 64;
        float mx = -3.4e38f;
        for (int i = 0; i < NS; ++i) mx = fmaxf(mx, attL[i * ATT_P + r] * smask[i]);
        float sum = 0.f;
        for (int i = 0; i < NS; ++i) {
            float e = expf(attL[i * ATT_P + r] * smask[i] - mx);
            awB[r * NS + i] = e;
            sum += e;
        }
        const float inv = 1.f / sum;
        for (int i = 0; i < NS; ++i) awB[r * NS + i] *= inv;
    }
    __syncthreads();

    float err1 = 0.f, err2 = 0.f;

    // ---- matmul 2: att_first(32x512) = awA(32x40) @ second(40x512); MSE vs first ----
    for (int t = wave; t < 64; t += 8) {
        const int mt = t >> 5;      // 0..1
        const int nt = t & 31;      // 0..31
        const int n  = nt * 16 + lm;
        const float* arow = awA + (mt * 16 + lm) * ATT_P;
        v8f c = {};
        for (int k = 0; k < ATT_P; k += 4) {
            const int kk = k + 2 * lh;
            v2f a = *(const v2f*)(arow + kk);
            v2f b;
            b.x = secondL[kk * D + n];
            b.y = secondL[(kk + 1) * D + n];
            c = __builtin_amdgcn_wmma_f32_16x16x4_f32(false, a, false, b, (short)0, c, false, false);
        }
#pragma unroll
        for (int i = 0; i < 8; ++i) {
            const int m = mt * 16 + lh * 8 + i;
            const float d = firstL[m * D + n] - c[i];
            err1 += d * d;
        }
    }

    // ---- matmul 3: att_second(48x512) = awB(48x32) @ first(32x512); MSE vs second ----
    // rows 36..47: awB rows zero -> c==0, secondL rows zero -> zero contribution.
    for (int t = wave; t < 96; t += 8) {
        const int mt = t / 32;      // 0..2
        const int nt = t % 32;      // 0..31
        const int n  = nt * 16 + lm;
        const float* arow = awB + (mt * 16 + lm) * NS;
        v8f c = {};
        for (int k = 0; k < NS; k += 4) {
            const int kk = k + 2 * lh;
            v2f a = *(const v2f*)(arow + kk);
            v2f b;
            b.x = firstL[kk * D + n];
            b.y = firstL[(kk + 1) * D + n];
            c = __builtin_amdgcn_wmma_f32_16x16x4_f32(false, a, false, b, (short)0, c, false, false);
        }
#pragma unroll
        for (int i = 0; i < 8; ++i) {
            const int m = mt * 16 + lh * 8 + i;
            const float d = secondL[m * D + n] - c[i];
            err2 += d * d;
        }
    }

    // ---- block reduce and write S[s,v] ----
    float acc = err1 * (1.f / (NS * D)) + err2 * (1.f / (NV * D));
    red[tid] = acc;
    __syncthreads();
    for (int st = 128; st > 0; st >>= 1) {
        if (tid < st) red[tid] += red[tid + st];
        __syncthreads();
    }
    if (tid == 0) S[s * NSAMP + v] = red[0];
}

__global__ void loss_kernel(const float* __restrict__ S, float* __restrict__ out)
{
    __shared__ float part[NSAMP];
    const int i = threadIdx.x;
    if (i < NSAMP) {
        const float diag = S[i * NSAMP + i];
        float mx = -3.4e38f;
        for (int j = 0; j < NSAMP; ++j) mx = fmaxf(mx, S[i * NSAMP + j]);
        float sum = 0.f;
        for (int j = 0; j < NSAMP; ++j) sum += expf(S[i * NSAMP + j] - mx);
        const float lse_r = mx + logf(sum);
        mx = -3.4e38f;
        for (int j = 0; j < NSAMP; ++j) mx = fmaxf(mx, S[j * NSAMP + i]);
        sum = 0.f;
        for (int j = 0; j < NSAMP; ++j) sum += expf(S[j * NSAMP + i] - mx);
        const float lse_c = mx + logf(sum);
        part[i] = (lse_r - diag) + (lse_c - diag);
    }
    __syncthreads();
    if (i == 0) {
        float a = 0.f;
        for (int j = 0; j < NSAMP; ++j) a += part[j];
        out[0] = a / (float)NSAMP;
    }
}

extern "C" void kernel_launch(void* const* d_in, const int* in_sizes, int n_in,
                              void* d_out, int out_size, void* d_ws, size_t ws_size,
                              hipStream_t stream)
{
    const float* span  = (const float*)d_in[0];
    const float* image = (const float*)d_in[1];
    const float* smask = (const float*)d_in[2];
    const float* imask = (const float*)d_in[3];
    float* S = (float*)d_ws;                       // 64x64 scores
    const size_t ldsBytes = LDS_FLOATS * sizeof(float);  // ~177 KB of 320 KB WGP LDS
    dim3 grid(NSAMP, NSAMP);
    pair_score_kernel<<<grid, 256, ldsBytes, stream>>>(span, image, smask, imask, S);
    loss_kernel<<<1, 64, 0, stream>>>(S, (float*)d_out);
}